// W_SpatialEmbLoss_15315853377947
// MI455X (gfx1250) — compile-verified
//
#include <hip/hip_runtime.h>
#include <math.h>

#define B_    8
#define CEMB  16
#define H_    512
#define W_    512
#define HW    (H_*W_)
#define NI    32
#define RADIUS_ 10
#define MAXNEI 10
#define BLKS_PER_B 64
#define PIX_PER_BLK (HW / BLKS_PER_B)     // 4096
#define NTHR 256
#define PIX_PER_THR (PIX_PER_BLK / NTHR)  // 16 contiguous pixels per thread

typedef __attribute__((ext_vector_type(2))) float v2f;
typedef __attribute__((ext_vector_type(8))) float v8f;

// ---------------- init workspace ----------------
__global__ void k_init(float* gsum, float* gcnt, int* gxmin, int* gxmax, int* gymin, int* gymax,
                       unsigned* presentT, float* intra, float* inter, float* focal, float* out) {
  int t = threadIdx.x;
  for (int i = t; i < B_*NI*CEMB; i += NTHR) gsum[i] = 0.f;
  for (int i = t; i < B_*NI; i += NTHR) {
    gcnt[i] = 0.f; gxmin[i] = 1<<30; gxmax[i] = -1; gymin[i] = 1<<30; gymax[i] = -1;
    presentT[i] = 0u;
  }
  if (t < B_) { intra[t] = 0.f; inter[t] = 0.f; }
  if (t == 0) { focal[0] = 0.f; out[0] = 0.f; }
}

// ---------------- pass 1: segment sums / counts / bboxes ----------------
__global__ void k_segstats(const float* __restrict__ emb, const int* __restrict__ inst,
                           float* gsum, float* gcnt, int* gxmin, int* gxmax, int* gymin, int* gymax) {
  __shared__ float lsum[NI*CEMB];
  __shared__ float lcnt[NI];
  __shared__ int lxmin[NI], lxmax[NI], lymin[NI], lymax[NI];
  int t = threadIdx.x;
  int b = blockIdx.x / BLKS_PER_B;
  int blk = blockIdx.x % BLKS_PER_B;
  for (int i = t; i < NI*CEMB; i += NTHR) lsum[i] = 0.f;
  if (t < NI) { lcnt[t]=0.f; lxmin[t]=1<<30; lxmax[t]=-1; lymin[t]=1<<30; lymax[t]=-1; }
  __syncthreads();
  const float* embB = emb + (size_t)b * CEMB * HW;
  const int*   instB = inst + (size_t)b * HW;
  int pbase = blk * PIX_PER_BLK + t * PIX_PER_THR;
  float acc[CEMB]; float cnt = 0.f;
  int cur = -1, xmn = 1<<30, xmx = -1, ymn = 1<<30, ymx = -1;
  for (int i = 0; i < PIX_PER_THR; ++i) {
    int p = pbase + i;
    int k = instB[p] & (NI - 1);
    int x = p & (W_ - 1), y = p >> 9;
    if (k != cur) {                    // flush run (instance maps are spatially coherent)
      if (cur >= 0) {
        for (int c = 0; c < CEMB; ++c) atomicAdd(&lsum[cur*CEMB + c], acc[c]);
        atomicAdd(&lcnt[cur], cnt);
        atomicMin(&lxmin[cur], xmn); atomicMax(&lxmax[cur], xmx);
        atomicMin(&lymin[cur], ymn); atomicMax(&lymax[cur], ymx);
      }
      cur = k; cnt = 0.f;
      for (int c = 0; c < CEMB; ++c) acc[c] = 0.f;
      xmn = 1<<30; xmx = -1; ymn = 1<<30; ymx = -1;
    }
    __builtin_prefetch(&embB[p + PIX_PER_THR], 0, 0);
    for (int c = 0; c < CEMB; ++c) acc[c] += embB[(size_t)c*HW + p];
    cnt += 1.f;
    xmn = min(xmn, x); xmx = max(xmx, x); ymn = min(ymn, y); ymx = max(ymx, y);
  }
  if (cur >= 0) {
    for (int c = 0; c < CEMB; ++c) atomicAdd(&lsum[cur*CEMB + c], acc[c]);
    atomicAdd(&lcnt[cur], cnt);
    atomicMin(&lxmin[cur], xmn); atomicMax(&lxmax[cur], xmx);
    atomicMin(&lymin[cur], ymn); atomicMax(&lymax[cur], ymx);
  }
  __syncthreads();
  for (int i = t; i < NI*CEMB; i += NTHR)
    if (lcnt[i / CEMB] > 0.f) atomicAdd(&gsum[(size_t)b*NI*CEMB + i], lsum[i]);
  if (t < NI && lcnt[t] > 0.f) {
    atomicAdd(&gcnt[b*NI + t], lcnt[t]);
    atomicMin(&gxmin[b*NI + t], lxmin[t]); atomicMax(&gxmax[b*NI + t], lxmax[t]);
    atomicMin(&gymin[b*NI + t], lymin[t]); atomicMax(&gymax[b*NI + t], lymax[t]);
  }
}

// ---------------- finalize centroids + expanded bboxes ----------------
__global__ void k_finalize(const float* gsum, const float* gcnt,
                           const int* gxmin, const int* gxmax, const int* gymin, const int* gymax,
                           float* keyn, float* normk, int* bx1, int* bx2, int* by1, int* by2) {
  int b = blockIdx.x, l = threadIdx.x;
  int idx = b*NI + l;
  float cnt = gcnt[idx];
  float key[CEMB]; float n2 = 0.f;
  for (int c = 0; c < CEMB; ++c) {
    float v = (cnt > 0.f) ? gsum[(size_t)idx*CEMB + c] / cnt : 0.f;
    key[c] = v; n2 += v*v;
  }
  float nrm = sqrtf(n2);
  float d = fmaxf(nrm, 1e-12f);
  for (int c = 0; c < CEMB; ++c) keyn[(size_t)idx*CEMB + c] = key[c] / d;
  normk[idx] = nrm / d;
  if (cnt > 0.f) {
    bx1[idx] = max(gxmin[idx] - RADIUS_, 0);
    bx2[idx] = min(gxmax[idx] + RADIUS_, W_);
    by1[idx] = max(gymin[idx] - RADIUS_, 0);
    by2[idx] = min(gymax[idx] + RADIUS_, H_);
  } else { bx1[idx] = 0; bx2[idx] = 0; by1[idx] = 0; by2[idx] = 0; }
}

// ---------------- pass 2: intra loss + bbox-presence bits ----------------
__global__ void k_intra_present(const float* __restrict__ emb, const int* __restrict__ inst,
                                const float* __restrict__ keyn, const float* __restrict__ normk,
                                const int* bx1, const int* bx2, const int* by1, const int* by2,
                                float* intra_acc, unsigned* presentT_g) {
  __shared__ float kf[NI*CEMB];
  __shared__ float nrm[NI];
  __shared__ int x1s[NI], x2s[NI], y1s[NI], y2s[NI];
  __shared__ unsigned pT[NI];      // pT[k] bit l: label k seen inside bbox of l
  __shared__ float red[NTHR];
  int t = threadIdx.x;
  int b = blockIdx.x / BLKS_PER_B;
  int blk = blockIdx.x % BLKS_PER_B;
  for (int i = t; i < NI*CEMB; i += NTHR) kf[i] = keyn[(size_t)b*NI*CEMB + i];
  if (t < NI) {
    nrm[t] = normk[b*NI + t];
    x1s[t] = bx1[b*NI+t]; x2s[t] = bx2[b*NI+t]; y1s[t] = by1[b*NI+t]; y2s[t] = by2[b*NI+t];
    pT[t] = 0u;
  }
  __syncthreads();
  const float* embB = emb + (size_t)b * CEMB * HW;
  const int*   instB = inst + (size_t)b * HW;
  int pbase = blk * PIX_PER_BLK + t * PIX_PER_THR;
  float lsum = 0.f;
  for (int i = 0; i < PIX_PER_THR; ++i) {
    int p = pbase + i;
    int k = instB[p] & (NI - 1);
    int x = p & (W_ - 1), y = p >> 9;
    __builtin_prefetch(&embB[p + PIX_PER_THR], 0, 0);
    float dot = 0.f, e2 = 0.f;
    for (int c = 0; c < CEMB; ++c) {
      float e = embB[(size_t)c*HW + p];
      dot += e * kf[k*CEMB + c];
      e2  += e * e;
    }
    float en = sqrtf(e2);
    float cosv = dot / (fmaxf(nrm[k], 1e-8f) * fmaxf(en, 1e-8f));
    lsum += expf(1.f - cosv) - 1.f;
    unsigned bits = 0u;
    for (int l = 0; l < NI; ++l)
      bits |= ((x >= x1s[l]) & (x < x2s[l]) & (y >= y1s[l]) & (y < y2s[l])) ? (1u << l) : 0u;
    atomicOr(&pT[k], bits);
  }
  red[t] = lsum; __syncthreads();
  for (int s = NTHR/2; s > 0; s >>= 1) { if (t < s) red[t] += red[t+s]; __syncthreads(); }
  if (t == 0) atomicAdd(&intra_acc[b], red[0]);
  if (t < NI && pT[t]) atomicOr(&presentT_g[b*NI + t], pT[t]);
}

// ---------------- focal loss ----------------
__global__ void k_focal(const float* __restrict__ seed, const int* __restrict__ labels, float* focal_acc) {
  __shared__ float red[NTHR];
  int t = threadIdx.x;
  size_t gid = (size_t)blockIdx.x * NTHR + t;
  size_t total = (size_t)B_ * HW;
  size_t stride = (size_t)gridDim.x * NTHR;
  float lsum = 0.f;
  for (size_t i = gid; i < total; i += stride) {
    size_t b = i / HW, p = i % HW;
    const float* sb = seed + b * 2 * HW;
    __builtin_prefetch(&sb[p + stride], 0, 0);
    float z0 = sb[p], z1 = sb[HW + p];
    int yv = labels[i];
    float m = fmaxf(z0, z1);
    float lse = m + logf(expf(z0 - m) + expf(z1 - m));
    float lpt = (yv ? z1 : z0) - lse;
    float pt = expf(lpt);
    float om = 1.f - pt;
    lsum += -(om * om) * lpt;
  }
  red[t] = lsum; __syncthreads();
  for (int s = NTHR/2; s > 0; s >>= 1) { if (t < s) red[t] += red[t+s]; __syncthreads(); }
  if (t == 0) atomicAdd(focal_acc, red[0]);
}

// ---------------- sim (WMMA f32 16x16x4) + mask + inter loss ----------------
__global__ void k_sim_inter(const float* __restrict__ keyn, const float* __restrict__ normk,
                            const unsigned* __restrict__ presentT, float* inter_val) {
  __shared__ float kf[NI*CEMB];
  __shared__ float nrm[NI];
  __shared__ unsigned pT[NI];
  __shared__ float simm[NI*NI];
  __shared__ float msk[NI*NI];
  __shared__ float redA[128], redB[128];
  int t = threadIdx.x;         // 128 threads = 4 waves of 32
  int b = blockIdx.x;
  for (int i = t; i < NI*CEMB; i += 128) kf[i] = keyn[(size_t)b*NI*CEMB + i];
  if (t < NI) { nrm[t] = normk[b*NI + t]; pT[t] = presentT[b*NI + t]; }
  __syncthreads();
  // sim = keyn (32x16) x keyn^T (16x32): 2x2 tiles of 16x16, K=16 as 4 steps of K=4
  int wave = t >> 5, lane = t & 31;
  int ti = (wave >> 1) * 16, tj = (wave & 1) * 16;
  int row = lane & 15;
  int sel = (lane < 16) ? 0 : 2;    // A 16x4 layout: lanes 0-15 carry K=+0/+1, 16-31 carry K=+2/+3
  v8f acc = {0.f,0.f,0.f,0.f,0.f,0.f,0.f,0.f};
  #pragma unroll
  for (int kk = 0; kk < CEMB; kk += 4) {
    v2f av, bv;
    av.x = kf[(ti + row)*CEMB + kk + sel];
    av.y = kf[(ti + row)*CEMB + kk + sel + 1];
    bv.x = kf[(tj + row)*CEMB + kk + sel];      // B[k][n] = keyn[tj+n][k] (column-major mirror)
    bv.y = kf[(tj + row)*CEMB + kk + sel + 1];
    acc = __builtin_amdgcn_wmma_f32_16x16x4_f32(false, av, false, bv, (short)0, acc, false, false);
  }
  int rbase = (lane < 16) ? 0 : 8;  // D layout: VGPR r holds rows r (lanes 0-15) / r+8 (lanes 16-31)
  #pragma unroll
  for (int r = 0; r < 8; ++r) {
    int i = ti + rbase + r, j = tj + row;
    float denom = fmaxf(nrm[i], 1e-8f) * fmaxf(nrm[j], 1e-8f);
    simm[i*NI + j] = fabsf(acc[r] / denom);
  }
  __syncthreads();
  // mask: top-10 largest nonzero present labels per row; row0/col0 = 0.5, [0,0] = 0
  if (t < NI) {
    int l = t;
    if (l == 0) {
      for (int k = 0; k < NI; ++k) msk[k] = 0.5f;
      msk[0] = 0.f;
    } else {
      msk[l*NI + 0] = 0.5f;
      int count = 0;
      for (int k = NI - 1; k >= 1; --k) {
        int pz = (pT[k] >> l) & 1;
        msk[l*NI + k] = (pz && count < MAXNEI) ? 1.f : 0.f;
        count += pz;
      }
    }
  }
  __syncthreads();
  float s = 0.f, ms = 0.f;
  for (int i = t; i < NI*NI; i += 128) {
    float m = msk[i];
    ms += m;
    s  += (expf(simm[i]) - 1.f) * m;
  }
  redA[t] = s; redB[t] = ms; __syncthreads();
  for (int st = 64; st > 0; st >>= 1) {
    if (t < st) { redA[t] += redA[t+st]; redB[t] += redB[t+st]; }
    __syncthreads();
  }
  if (t == 0) inter_val[b] = (redB[0] > 0.f) ? redA[0] / fmaxf(redB[0], 1e-12f) : 0.f;
}

// ---------------- combine ----------------
__global__ void k_combine(const float* intra_acc, const float* inter_val, const float* focal_acc, float* out) {
  if (threadIdx.x == 0 && blockIdx.x == 0) {
    float si = 0.f, se = 0.f;
    for (int b = 0; b < B_; ++b) { si += intra_acc[b] / (float)HW; se += inter_val[b]; }
    float focal = focal_acc[0] / (float)((size_t)B_ * HW);
    out[0] = focal * 1.0f + (se / (float)B_) * 1.0f + (si / (float)B_) * 10.0f;  // W_SEED, W_INST, W_VAR
  }
}

extern "C" void kernel_launch(void* const* d_in, const int* in_sizes, int n_in,
                              void* d_out, int out_size, void* d_ws, size_t ws_size,
                              hipStream_t stream) {
  const float* seed   = (const float*)d_in[0];   // [8,2,512,512]
  const float* emb    = (const float*)d_in[1];   // [8,16,512,512]
  const int*   labels = (const int*)d_in[2];     // [8,512,512]
  const int*   inst   = (const int*)d_in[3];     // [8,1,512,512]
  float* out = (float*)d_out;

  // workspace carve-up
  float* gsum  = (float*)d_ws;                       // 8*32*16
  float* gcnt  = gsum + B_*NI*CEMB;                  // 8*32
  int* gxmin   = (int*)(gcnt + B_*NI);
  int* gxmax   = gxmin + B_*NI;
  int* gymin   = gxmax + B_*NI;
  int* gymax   = gymin + B_*NI;
  float* keyn  = (float*)(gymax + B_*NI);            // 8*32*16
  float* normk = keyn + B_*NI*CEMB;                  // 8*32
  int* bx1     = (int*)(normk + B_*NI);
  int* bx2     = bx1 + B_*NI;
  int* by1     = bx2 + B_*NI;
  int* by2     = by1 + B_*NI;
  unsigned* presentT = (unsigned*)(by2 + B_*NI);     // 8*32
  float* intra = (float*)(presentT + B_*NI);         // 8
  float* inter = intra + B_;                         // 8
  float* focal = inter + B_;                         // 1

  k_init<<<1, NTHR, 0, stream>>>(gsum, gcnt, gxmin, gxmax, gymin, gymax,
                                 presentT, intra, inter, focal, out);
  k_segstats<<<B_*BLKS_PER_B, NTHR, 0, stream>>>(emb, inst, gsum, gcnt,
                                                 gxmin, gxmax, gymin, gymax);
  k_finalize<<<B_, NI, 0, stream>>>(gsum, gcnt, gxmin, gxmax, gymin, gymax,
                                    keyn, normk, bx1, bx2, by1, by2);
  k_intra_present<<<B_*BLKS_PER_B, NTHR, 0, stream>>>(emb, inst, keyn, normk,
                                                      bx1, bx2, by1, by2, intra, presentT);
  k_focal<<<512, NTHR, 0, stream>>>(seed, labels, focal);
  k_sim_inter<<<B_, 128, 0, stream>>>(keyn, normk, presentT, inter);
  k_combine<<<1, 32, 0, stream>>>(intra, inter, focal, out);
}